// GraphAttentionLayer_58428735095220
// MI455X (gfx1250) — compile-verified
//
#include <hip/hip_runtime.h>
#include <hip/hip_bf16.h>
#include <math.h>

#define N_NODES 16384
#define N_EDGES 524288
#define DD 128
#define LDSK 132   // padded k-stride: 528 B -> 4-bank shift per column, conflict-free
#define ALPHA_SLOPE 0.2f

typedef float v2f __attribute__((ext_vector_type(2)));
typedef float v8f __attribute__((ext_vector_type(8)));

// ---------------------------------------------------------------------------
// H = X @ W   [16384,128] x [128,128], f32, via V_WMMA_F32_16X16X4_F32.
// Block: 256 threads (8 waves). Each block covers 128 rows; each wave a
// 16-row strip across all 128 output columns (8 accumulator tiles).
// W is staged TRANSPOSED in LDS: BsT[n*LDSK + k], so a B fragment
// (B[k][n], B[k+1][n]) is one aligned ds_load_b64. A fragments are
// software-pipelined one k-step ahead to hide global latency behind WMMA.
// ---------------------------------------------------------------------------
__global__ __launch_bounds__(256) void gat_gemm_h(const float* __restrict__ X,
                                                  const float* __restrict__ W,
                                                  float* __restrict__ H) {
  __shared__ float BsT[DD * LDSK];  // ~66 KB of 320 KB WGP LDS
  const int tid  = threadIdx.x;
  const int lane = tid & 31;
  const int wave = tid >> 5;

  // cooperative transpose-load of W into LDS (coalesced global reads)
  for (int i = tid; i < DD * DD; i += 256) {
    const int k = i >> 7;        // row of W (K index)
    const int n = i & (DD - 1);  // col of W (N index)
    BsT[n * LDSK + k] = W[i];
  }
  __syncthreads();

  const int row0 = blockIdx.x * 128 + wave * 16;
  const int m    = lane & 15;          // M index within tile
  const int khi  = (lane >> 4) << 1;   // K sub-offset: lanes 16-31 hold K+2

  v8f acc[8] = {};

  const float* Aptr = X + (size_t)(row0 + m) * DD + khi;

  // issue the 8 WMMAs for one k-step (aF spans K = kk+khi .. kk+khi+1)
  auto wmma_step = [&](v2f aF, int kk) {
#pragma unroll
    for (int t = 0; t < 8; ++t) {
      const int n = t * 16 + m;
      v2f bF = *(const v2f*)(BsT + n * LDSK + kk + khi);  // ds_load_b64
      acc[t] = __builtin_amdgcn_wmma_f32_16x16x4_f32(
          /*neg_a=*/false, aF, /*neg_b=*/false, bF,
          /*c_mod=*/(short)0, acc[t], /*reuse_a=*/false, /*reuse_b=*/false);
    }
  };

  // software pipeline: A fragment for k-step kk+4 loads while kk computes
  v2f av = *(const v2f*)(Aptr);
#pragma unroll 4
  for (int kk = 0; kk < DD - 4; kk += 4) {
    __builtin_prefetch(Aptr + kk + 36, 0, 1);          // global_prefetch ahead
    v2f av_next = *(const v2f*)(Aptr + kk + 4);
    wmma_step(av, kk);
    av = av_next;
  }
  wmma_step(av, DD - 4);

  // C/D layout: lane L, VGPR v -> M = v + 8*(L>=16), N = L%16
  const int mhi = (lane >> 4) << 3;
#pragma unroll
  for (int t = 0; t < 8; ++t)
#pragma unroll
    for (int v = 0; v < 8; ++v)
      H[(size_t)(row0 + mhi + v) * DD + t * 16 + m] = acc[t][v];
}

// ---------------------------------------------------------------------------
// Per-node attention scalars: alpha[i] = ( h[i]·a[0:128], h[i]·a[128:256] )
// One wave32 per node, float4 loads, shfl_xor reduction.
// ---------------------------------------------------------------------------
__global__ __launch_bounds__(256) void gat_alpha(const float* __restrict__ H,
                                                 const float* __restrict__ Avec,
                                                 float2* __restrict__ alpha) {
  const int node = blockIdx.x * 8 + (threadIdx.x >> 5);
  const int lane = threadIdx.x & 31;
  const float4 h4 = ((const float4*)(H + (size_t)node * DD))[lane];
  const float4 a1 = ((const float4*)Avec)[lane];
  const float4 a2 = ((const float4*)(Avec + DD))[lane];
  float s1 = h4.x * a1.x + h4.y * a1.y + h4.z * a1.z + h4.w * a1.w;
  float s2 = h4.x * a2.x + h4.y * a2.y + h4.z * a2.z + h4.w * a2.w;
  for (int off = 16; off > 0; off >>= 1) {
    s1 += __shfl_xor(s1, off, 32);
    s2 += __shfl_xor(s2, off, 32);
  }
  if (lane == 0) alpha[node] = make_float2(s1, s2);
}

// ---------------------------------------------------------------------------
__global__ void gat_init(float* __restrict__ m, float* __restrict__ Z,
                         int* __restrict__ deg, float* __restrict__ S) {
  const int i = blockIdx.x * blockDim.x + threadIdx.x;
  if (i < N_NODES) { m[i] = 0.0f; Z[i] = 0.0f; deg[i] = 0; }
  if (i < DD) S[i] = 0.0f;
}

__device__ __forceinline__ float edge_score(const float2* alpha, int r, int c) {
  float v = alpha[r].x + alpha[c].y;
  return v > 0.0f ? v : ALPHA_SLOPE * v;
}

// Pass 1: row max (candidates >= 0, init +0.0 -> int atomicMax is exact) + degree
__global__ void gat_edge_max(const long long* __restrict__ ei,
                             const float2* __restrict__ alpha,
                             float* __restrict__ m, int* __restrict__ deg) {
  const int e = blockIdx.x * blockDim.x + threadIdx.x;
  if (e >= N_EDGES) return;
  const int r = (int)ei[e];
  const int c = (int)ei[N_EDGES + e];
  const float v = edge_score(alpha, r, c);
  if (v > 0.0f) atomicMax((int*)(m + r), __float_as_int(v));
  atomicAdd(deg + r, 1);
}

// Pass 2: Z[row] += exp(e - m[row])
__global__ void gat_edge_sum(const long long* __restrict__ ei,
                             const float2* __restrict__ alpha,
                             const float* __restrict__ m, float* __restrict__ Z) {
  const int e = blockIdx.x * blockDim.x + threadIdx.x;
  if (e >= N_EDGES) return;
  const int r = (int)ei[e];
  const int c = (int)ei[N_EDGES + e];
  const float v = edge_score(alpha, r, c);
  atomicAdd(Z + r, __expf(v - m[r]));
}

// Z_i -> 1 / (Z_i + (N - deg_i) e^{-m_i});  base_i = e^{-m_i} * invZ_i
__global__ void gat_finalize(const float* __restrict__ m, const int* __restrict__ deg,
                             float* __restrict__ Z, float* __restrict__ base) {
  const int i = blockIdx.x * blockDim.x + threadIdx.x;
  if (i >= N_NODES) return;
  const float em  = __expf(-m[i]);
  const float z   = Z[i] + (float)(N_NODES - deg[i]) * em;
  const float inv = 1.0f / z;
  Z[i]    = inv;
  base[i] = em * inv;
}

// S[c] = sum_i H[i][c]; block = 128 threads (one per col), 128 rows per block
__global__ __launch_bounds__(128) void gat_colsum(const float* __restrict__ H,
                                                  float* __restrict__ S) {
  const int col = threadIdx.x;
  const int r0  = blockIdx.x * 128;
  float s = 0.0f;
  for (int r = 0; r < 128; ++r) s += H[(size_t)(r0 + r) * DD + col];
  atomicAdd(S + col, s);
}

// out[i][c] = base[i] * S[c]   (full overwrite of poisoned d_out)
__global__ void gat_out_base(const float* __restrict__ base, const float* __restrict__ S,
                             float* __restrict__ out) {
  const int idx = blockIdx.x * blockDim.x + threadIdx.x;
  out[idx] = base[idx >> 7] * S[idx & (DD - 1)];
}

// out[row] += ((e^{e-m} - e^{-m}) * invZ) * h[col] ; one wave32 per edge
__global__ __launch_bounds__(256) void gat_edge_scatter(
    const long long* __restrict__ ei, const float2* __restrict__ alpha,
    const float* __restrict__ m, const float* __restrict__ invZ,
    const float* __restrict__ H, float* __restrict__ out) {
  const int gt   = blockIdx.x * blockDim.x + threadIdx.x;
  const int e    = gt >> 5;
  const int lane = gt & 31;
  const int r = (int)ei[e];
  const int c = (int)ei[N_EDGES + e];
  const float v  = edge_score(alpha, r, c);
  const float mm = m[r];
  const float w  = (__expf(v - mm) - __expf(-mm)) * invZ[r];
  const float4 h4 = ((const float4*)(H + (size_t)c * DD))[lane];
  float* o = out + (size_t)r * DD + lane * 4;
  atomicAdd(o + 0, w * h4.x);
  atomicAdd(o + 1, w * h4.y);
  atomicAdd(o + 2, w * h4.z);
  atomicAdd(o + 3, w * h4.w);
}

// ---------------------------------------------------------------------------
extern "C" void kernel_launch(void* const* d_in, const int* in_sizes, int n_in,
                              void* d_out, int out_size, void* d_ws, size_t ws_size,
                              hipStream_t stream) {
  const float*     X    = (const float*)d_in[0];       // [N,128]
  const long long* EI   = (const long long*)d_in[1];   // [2,E] int64
  const float*     W    = (const float*)d_in[2];       // [128,128]
  const float*     Avec = (const float*)d_in[3];       // [256]
  float* out = (float*)d_out;                          // [N,128]

  char* ws = (char*)d_ws;
  float*  H     = (float*)ws;  ws += sizeof(float)  * N_NODES * DD;
  float2* alpha = (float2*)ws; ws += sizeof(float2) * N_NODES;
  float*  m     = (float*)ws;  ws += sizeof(float)  * N_NODES;
  float*  Z     = (float*)ws;  ws += sizeof(float)  * N_NODES;   // sum -> invZ
  float*  base  = (float*)ws;  ws += sizeof(float)  * N_NODES;
  int*    deg   = (int*)ws;    ws += sizeof(int)    * N_NODES;
  float*  S     = (float*)ws;  ws += sizeof(float)  * DD;

  gat_init        <<<(N_NODES + 255) / 256, 256, 0, stream>>>(m, Z, deg, S);
  gat_gemm_h      <<<N_NODES / 128,        256, 0, stream>>>(X, W, H);
  gat_alpha       <<<N_NODES / 8,          256, 0, stream>>>(H, Avec, alpha);
  gat_colsum      <<<N_NODES / 128,        128, 0, stream>>>(H, S);
  gat_edge_max    <<<N_EDGES / 256,        256, 0, stream>>>(EI, alpha, m, deg);
  gat_edge_sum    <<<N_EDGES / 256,        256, 0, stream>>>(EI, alpha, m, Z);
  gat_finalize    <<<(N_NODES + 255) / 256,256, 0, stream>>>(m, deg, Z, base);
  gat_out_base    <<<N_NODES * DD / 256,   256, 0, stream>>>(base, S, out);
  gat_edge_scatter<<<N_EDGES * 32 / 256,   256, 0, stream>>>(EI, alpha, m, Z, H, out);
}